// CRFLoss_87110526697618
// MI455X (gfx1250) — compile-verified
//
#include <hip/hip_runtime.h>

typedef __attribute__((ext_vector_type(2))) float v2f;
typedef __attribute__((ext_vector_type(8))) float v8f;
typedef int v4i_vs __attribute__((vector_size(16)));   // b128 payload type per clang diag

#define KK 64
#define LSTRIDE 68          // 64 + 4 pad: keeps 16B alignment, staggers LDS banks per row
#define START_IDX 62
#define END_IDX 63
#define NEG_BIG (-3.402823466e38f)

// --- CDNA5 async global->LDS path (gated; fallback = VGPR staging) ---------
#if defined(__has_builtin)
#  if __has_builtin(__builtin_amdgcn_global_load_async_to_lds_b128)
#    define HAVE_ASYNC_LDS 1
#  endif
#  if __has_builtin(__builtin_amdgcn_s_wait_asynccnt)
#    define HAVE_WAIT_ASYNC_BUILTIN 1
#  endif
#endif

#if defined(HAVE_ASYNC_LDS)
#define AS_GLOBAL __attribute__((address_space(1)))
#define AS_LDS    __attribute__((address_space(3)))
__device__ __forceinline__ void async_b128(const float* gsrc, float* lds) {
  // copies 16 bytes per lane from the global address gsrc to the LDS address lds, tracked by ASYNCcnt
  __builtin_amdgcn_global_load_async_to_lds_b128(
      (AS_GLOBAL v4i_vs*)gsrc, (AS_LDS v4i_vs*)lds, 0, 0);
}
__device__ __forceinline__ void wait_async0() {
#if defined(HAVE_WAIT_ASYNC_BUILTIN)
  __builtin_amdgcn_s_wait_asynccnt(0);
#else
  asm volatile("s_wait_asynccnt 0" ::: "memory");
#endif
}
#endif

// One workgroup per batch element (memory-bound: 4 MB/seq @ 23.3 TB/s).
// Per step: async-DMA next 64x64 f32 matrix into the other LDS buffer,
// shuffle-reduce the global max, in-place exp, then waves 0..3 each reduce
// one 16-column tile via V_WMMA_F32_16X16X4_F32 with an all-ones A matrix.
__launch_bounds__(256)
__global__ void crf_forward_kernel(const float* __restrict__ scores,
                                   const long long* __restrict__ targets,
                                   const long long* __restrict__ lengths,
                                   float* __restrict__ partials,  // [2*B]
                                   int T) {
  __shared__ __align__(16) float buf[2][KK * LSTRIDE];
  __shared__ float fs[KK];
  __shared__ float red[256];   // gold reduction tree; red[0..7] reused for per-wave maxima

  const int b   = blockIdx.x;
  const int tid = threadIdx.x;
  const int wv  = tid >> 5;    // wave id 0..7 (wave32)
  int len = (int)lengths[b];
  if (len > T) len = T;

  const float* base = scores + (size_t)b * T * (KK * KK);

  // ---- gold-path partial sum (valid steps t < len) ----
  float g = 0.0f;
  for (int tt = tid; tt < T; tt += 256) {
    if (tt < len) {
      int tg = (int)targets[(size_t)b * T + tt];
      g += base[(size_t)tt * (KK * KK) + tg];
    }
  }
  red[tid] = g;
  __syncthreads();
  for (int s = 128; s > 0; s >>= 1) {
    if (tid < s) red[tid] += red[tid + s];
    __syncthreads();
  }
  if (tid == 0) partials[b] = red[0];

  // ---- forward-state init: fs = scores[b,0,START_IDX,:] ----
  if (tid < KK) fs[tid] = base[START_IDX * KK + tid];

  // Each thread owns 16 contiguous floats (one quarter-row) of the matrix.
  const int row = tid >> 2;          // 0..63
  const int c0  = (tid & 3) * 16;    // 0,16,32,48

#if !defined(HAVE_ASYNC_LDS)
  float4 r0, r1, r2, r3;
#endif

  if (len > 1) {  // preload matrix for t=1 into buf[0]
    const float* gm = base + (size_t)(KK * KK) + row * KK + c0;
    float*       dm = &buf[0][row * LSTRIDE + c0];
#if defined(HAVE_ASYNC_LDS)
    async_b128(gm,      dm);
    async_b128(gm + 4,  dm + 4);
    async_b128(gm + 8,  dm + 8);
    async_b128(gm + 12, dm + 12);
    wait_async0();
#else
    const float4* m = (const float4*)gm;
    float4*       d = (float4*)dm;
    d[0] = m[0]; d[1] = m[1]; d[2] = m[2]; d[3] = m[3];
#endif
  }
  __syncthreads();

  for (int t = 1; t < len; ++t) {
    const int cur = (t - 1) & 1;
    const int nxt = t & 1;
    const bool havnext = (t + 1) < len;

    if (havnext) {  // start global->LDS DMA for t+1; overlaps all compute below
      const float* gm = base + (size_t)(t + 1) * (KK * KK) + row * KK + c0;
#if defined(HAVE_ASYNC_LDS)
      float* dm = &buf[nxt][row * LSTRIDE + c0];
      async_b128(gm,      dm);
      async_b128(gm + 4,  dm + 4);
      async_b128(gm + 8,  dm + 8);
      async_b128(gm + 12, dm + 12);
#else
      const float4* m = (const float4*)gm;
      r0 = m[0]; r1 = m[1]; r2 = m[2]; r3 = m[3];
#endif
      if (t + 2 < len)
        __builtin_prefetch(base + (size_t)(t + 2) * (KK * KK) + row * KK + c0, 0, 0);
    }

    // ---- pass 1: M = max_{i,j}(s[i,j] + fs[i])  (=> all exp args <= 0) ----
    const float fsr = fs[row];
    float* p = &buf[cur][row * LSTRIDE + c0];
    float lm = NEG_BIG;
#pragma unroll
    for (int k = 0; k < 16; ++k) lm = fmaxf(lm, p[k] + fsr);
#pragma unroll
    for (int off = 16; off > 0; off >>= 1)
      lm = fmaxf(lm, __shfl_xor(lm, off, 32));
    if ((tid & 31) == 0) red[wv] = lm;
    __syncthreads();
    float M = red[0];
#pragma unroll
    for (int w = 1; w < 8; ++w) M = fmaxf(M, red[w]);

    // ---- pass 2: in-place E[i,j] = exp(s[i,j] + fs[i] - M) ----
#pragma unroll
    for (int k = 0; k < 16; ++k) p[k] = __expf(p[k] + fsr - M);
    __syncthreads();

    // ---- waves 0..3: tile column sums via WMMA f32 16x16x4, A = ones ----
    // D[m,n] = sum_k 1 * E[k][n]: every row of D is the column sum, so the
    // result is invariant to the exact K-row/lane mapping of the B operand.
    if (wv < 4) {
      const int lane = tid & 31;
      const int half = lane >> 4;   // 0 | 1
      const int ln   = lane & 15;
      const int j0   = wv * 16;     // this wave's 16-column tile
      v2f ones; ones[0] = 1.0f; ones[1] = 1.0f;
      v8f acc = {};
      const float* E = &buf[cur][0];
#pragma unroll
      for (int kkb = 0; kkb < 16; ++kkb) {
        const int rb = kkb * 4 + half;
        v2f bm;
        bm[0] = E[rb * LSTRIDE + j0 + ln];
        bm[1] = E[(rb + 2) * LSTRIDE + j0 + ln];
        acc = __builtin_amdgcn_wmma_f32_16x16x4_f32(false, ones, false, bm,
                                                    (short)0, acc, false, false);
      }
      // D row 0 lives in accumulator VGPR0 (component 0), lanes 0..15, N=lane.
      if (half == 0) fs[j0 + ln] = M + __logf(acc[0]);
    }

#if defined(HAVE_ASYNC_LDS)
    if (havnext) wait_async0();   // buf[nxt] fully landed before hand-off
    __syncthreads();
#else
    __syncthreads();
    if (havnext) {
      float4* d = (float4*)(&buf[nxt][row * LSTRIDE + c0]);
      d[0] = r0; d[1] = r1; d[2] = r2; d[3] = r3;
    }
    __syncthreads();
#endif
  }

  if (tid == 0) partials[gridDim.x + b] = fs[END_IDX];
}

// Deterministic final reduction: out = (sum(fs_end) - sum(gold)) / B
__global__ void crf_reduce_kernel(const float* __restrict__ partials,
                                  float* __restrict__ out, int B) {
  __shared__ float rg[128];
  __shared__ float rf[128];
  const int tid = threadIdx.x;
  float pg = 0.0f, pf = 0.0f;
  for (int i = tid; i < B; i += 128) {
    pg += partials[i];
    pf += partials[B + i];
  }
  rg[tid] = pg; rf[tid] = pf;
  __syncthreads();
  for (int s = 64; s > 0; s >>= 1) {
    if (tid < s) { rg[tid] += rg[tid + s]; rf[tid] += rf[tid + s]; }
    __syncthreads();
  }
  if (tid == 0) out[0] = (rf[0] - rg[0]) / (float)B;
}

extern "C" void kernel_launch(void* const* d_in, const int* in_sizes, int n_in,
                              void* d_out, int out_size, void* d_ws, size_t ws_size,
                              hipStream_t stream) {
  const float*     scores  = (const float*)d_in[0];
  const long long* targets = (const long long*)d_in[1];
  const long long* lengths = (const long long*)d_in[2];

  const int B = in_sizes[2];              // lengths: (B,)
  const int T = in_sizes[1] / B;          // targets: (B,T)

  float* partials = (float*)d_ws;         // [2*B] floats of scratch

  crf_forward_kernel<<<B, 256, 0, stream>>>(scores, targets, lengths, partials, T);
  crf_reduce_kernel<<<1, 128, 0, stream>>>(partials, (float*)d_out, B);
}